// DynamicGraphLearner_48490180772619
// MI455X (gfx1250) — compile-verified
//
#include <hip/hip_runtime.h>
#include <math.h>

// ---------------------------------------------------------------------------
// DynamicGraphLearner fused pipeline for MI455X (gfx1250, wave32).
//
// Roofline: ~140MB of mandatory traffic @ 23.3TB/s ~= 6us; compute is ~0.4
// GFLOP (rank-2 + rank-10 scores) so the kernel set is built around streaming
// base_adj once from HBM and exploiting the 192MB L2 for the repeated softmax
// passes. The two low-rank GEMMs are done with V_WMMA_F32_16X16X4_F32 tiles.
// ---------------------------------------------------------------------------

#define NN 4096
#define TOPK 20
#define EPS_F 1e-6f
#define ETA_F 0.60653065971263342f   // exp(-0.5)
#define NSAMP 192                    // 16*12

typedef float v2f __attribute__((ext_vector_type(2)));
typedef float v8f __attribute__((ext_vector_type(8)));

// ---------------- ws zeroing (feat acc 2N, sev acc N, flag) ----------------
__global__ void dg_kZ(float* ws) {
    int t = blockIdx.x * blockDim.x + threadIdx.x;
    if (t < 3 * NN + 256) ws[t] = 0.0f;
}

// ---------------- partial sums over the (16*12) sample axis ----------------
__global__ __launch_bounds__(256) void dg_kP1(const float* __restrict__ xh,
                                              const float* __restrict__ ev,
                                              float* __restrict__ facc,
                                              float* __restrict__ sacc) {
    int n  = blockIdx.x * 256 + threadIdx.x;   // node
    int s0 = blockIdx.y * (NSAMP / 8);         // sample chunk
    float a0 = 0.f, a1 = 0.f, sv = 0.f;
    for (int s = s0; s < s0 + (NSAMP / 8); ++s) {
        size_t base = ((size_t)s * NN + n);
        a0 += xh[base * 2 + 0];
        a1 += xh[base * 2 + 1];
        sv += ev[base];
    }
    atomicAdd(&facc[2 * n + 0], a0);
    atomicAdd(&facc[2 * n + 1], a1);
    atomicAdd(&sacc[n], sv);
}

// ---------------- finalize feat (L2-normalized) / sev / impacted flag ------
__global__ __launch_bounds__(256) void dg_kP2(float* feat, float* sevp, int* flag) {
    int n = blockIdx.x * 256 + threadIdx.x;
    float f0 = feat[2 * n + 0] * (1.f / NSAMP);
    float f1 = feat[2 * n + 1] * (1.f / NSAMP);
    float nr = sqrtf(f0 * f0 + f1 * f1) + EPS_F;
    feat[2 * n + 0] = f0 / nr;
    feat[2 * n + 1] = f1 / nr;
    float sv = sevp[n] * (1.f / NSAMP);
    sevp[n] = sv;
    if (sv > 0.f) atomicOr(flag, 1);
}

// ---------------- main fused WMMA kernel: writes A_dyn (pre-topk) ----------
// One workgroup (8 waves) owns 16 rows. Three passes over 256 column tiles:
//   A: per-row maxes of the three relu'd score matrices
//   B: per-row sum(exp(score-max))
//   C: fused A_dyn value -> d_out
// Scores are recomputed each pass (base_adj row stripe is L2-resident).
__global__ __launch_bounds__(256) void dg_kW(const float* __restrict__ base,
                                             const float* __restrict__ nv1,
                                             const float* __restrict__ nv2,
                                             const float* __restrict__ feat,
                                             const float* __restrict__ sev,
                                             const int*   __restrict__ flagp,
                                             const float* __restrict__ alphap,
                                             float* __restrict__ out) {
    const int tid  = threadIdx.x;
    const int wave = tid >> 5;
    const int lane = tid & 31;
    const int half = lane >> 4;   // C-fragment half: rows v+8*half
    const int c    = lane & 15;   // column within 16-wide tile / row within A
    const int i0   = blockIdx.x * 16;

    __shared__ float s_red[8][3][16];   // per-wave partials [wave][score][row]
    __shared__ float s_fin[2][3][16];   // 0 = max, 1 = sum

    // ---- A fragments (constant for the whole stripe) ----
    // A is 16x4 (K padded): VGPR k-pairs (2*half+0, 2*half+1) within 4-chunk.
    v2f a_h, a_adp0, a_adp1, a_adp2;
    {
        int row = i0 + c;
        if (half == 0) { a_h[0] = feat[2 * row]; a_h[1] = feat[2 * row + 1]; }
        else           { a_h[0] = 0.f;           a_h[1] = 0.f; }
        int k0;
        k0 = 0 + 2 * half;
        a_adp0[0] = nv1[row * 10 + k0];
        a_adp0[1] = nv1[row * 10 + k0 + 1];
        k0 = 4 + 2 * half;
        a_adp1[0] = nv1[row * 10 + k0];
        a_adp1[1] = nv1[row * 10 + k0 + 1];
        k0 = 8 + 2 * half;                      // K = 8,9 valid; 10,11 pad
        a_adp2[0] = (k0     < 10) ? nv1[row * 10 + k0]     : 0.f;
        a_adp2[1] = (k0 + 1 < 10) ? nv1[row * 10 + k0 + 1] : 0.f;
    }
    float sevr[8];
    #pragma unroll
    for (int v = 0; v < 8; ++v) sevr[v] = sev[i0 + v + 8 * half];

    const float gamma = 1.f / (1.f + __expf(-alphap[0]));
    const float eflag = (*flagp != 0) ? 1.f : 0.f;

    // ---- tile score evaluator (4 WMMAs per call) ----
    auto tile_scores = [&](int j0, v8f& ch, v8f& ca, float* ev, float* bv) {
        v2f b_h, b0, b1, b2;
        int col = j0 + c;
        if (half == 0) { b_h[0] = feat[2 * col]; b_h[1] = feat[2 * col + 1]; }
        else           { b_h[0] = 0.f;           b_h[1] = 0.f; }
        int k0;
        k0 = 0 + 2 * half;
        b0[0] = nv2[(size_t)k0 * NN + col];
        b0[1] = nv2[(size_t)(k0 + 1) * NN + col];
        k0 = 4 + 2 * half;
        b1[0] = nv2[(size_t)k0 * NN + col];
        b1[1] = nv2[(size_t)(k0 + 1) * NN + col];
        k0 = 8 + 2 * half;
        b2[0] = (k0     < 10) ? nv2[(size_t)k0 * NN + col]       : 0.f;
        b2[1] = (k0 + 1 < 10) ? nv2[(size_t)(k0 + 1) * NN + col] : 0.f;

        v8f z = {0.f, 0.f, 0.f, 0.f, 0.f, 0.f, 0.f, 0.f};
        ch = __builtin_amdgcn_wmma_f32_16x16x4_f32(false, a_h, false, b_h,
                                                   (short)0, z, false, false);
        ca = __builtin_amdgcn_wmma_f32_16x16x4_f32(false, a_adp0, false, b0,
                                                   (short)0, z, false, false);
        ca = __builtin_amdgcn_wmma_f32_16x16x4_f32(false, a_adp1, false, b1,
                                                   (short)0, ca, false, false);
        ca = __builtin_amdgcn_wmma_f32_16x16x4_f32(false, a_adp2, false, b2,
                                                   (short)0, ca, false, false);
        #pragma unroll
        for (int v = 0; v < 8; ++v) {
            ch[v] = fmaxf(ch[v], 0.f);                       // relu(history)
            ca[v] = fmaxf(ca[v], 0.f);                       // relu(adaptive)
            float b = base[(size_t)(i0 + v + 8 * half) * NN + j0 + c];
            bv[v] = b;
            ev[v] = fmaxf(sevr[v] * ETA_F * b, 0.f);         // relu(event)
        }
    };

    // ================= pass A: row maxes =================
    float mh[8], me[8], ma[8];
    #pragma unroll
    for (int v = 0; v < 8; ++v) { mh[v] = 0.f; me[v] = 0.f; ma[v] = 0.f; }
    for (int jt = wave; jt < NN / 16; jt += 8) {
        v8f ch, ca; float ev[8], bv[8];
        tile_scores(jt * 16, ch, ca, ev, bv);
        #pragma unroll
        for (int v = 0; v < 8; ++v) {
            mh[v] = fmaxf(mh[v], ch[v]);
            ma[v] = fmaxf(ma[v], ca[v]);
            me[v] = fmaxf(me[v], ev[v]);
        }
    }
    #pragma unroll
    for (int v = 0; v < 8; ++v)
        for (int off = 1; off < 16; off <<= 1) {
            mh[v] = fmaxf(mh[v], __shfl_xor(mh[v], off, 16));
            me[v] = fmaxf(me[v], __shfl_xor(me[v], off, 16));
            ma[v] = fmaxf(ma[v], __shfl_xor(ma[v], off, 16));
        }
    if (c == 0)
        #pragma unroll
        for (int v = 0; v < 8; ++v) {
            int r = v + 8 * half;
            s_red[wave][0][r] = mh[v];
            s_red[wave][1][r] = me[v];
            s_red[wave][2][r] = ma[v];
        }
    __syncthreads();
    if (tid < 48) {
        int s = tid >> 4, r = tid & 15;
        float m = s_red[0][s][r];
        for (int w = 1; w < 8; ++w) m = fmaxf(m, s_red[w][s][r]);
        s_fin[0][s][r] = m;
    }
    __syncthreads();
    float gmh[8], gme[8], gma[8];
    #pragma unroll
    for (int v = 0; v < 8; ++v) {
        int r = v + 8 * half;
        gmh[v] = s_fin[0][0][r];
        gme[v] = s_fin[0][1][r];
        gma[v] = s_fin[0][2][r];
    }

    // ================= pass B: row sum-exp =================
    float sh[8], se[8], sa[8];
    #pragma unroll
    for (int v = 0; v < 8; ++v) { sh[v] = 0.f; se[v] = 0.f; sa[v] = 0.f; }
    for (int jt = wave; jt < NN / 16; jt += 8) {
        v8f ch, ca; float ev[8], bv[8];
        tile_scores(jt * 16, ch, ca, ev, bv);
        #pragma unroll
        for (int v = 0; v < 8; ++v) {
            sh[v] += __expf(ch[v] - gmh[v]);
            se[v] += __expf(ev[v] - gme[v]);
            sa[v] += __expf(ca[v] - gma[v]);
        }
    }
    #pragma unroll
    for (int v = 0; v < 8; ++v)
        for (int off = 1; off < 16; off <<= 1) {
            sh[v] += __shfl_xor(sh[v], off, 16);
            se[v] += __shfl_xor(se[v], off, 16);
            sa[v] += __shfl_xor(sa[v], off, 16);
        }
    if (c == 0)
        #pragma unroll
        for (int v = 0; v < 8; ++v) {
            int r = v + 8 * half;
            s_red[wave][0][r] = sh[v];
            s_red[wave][1][r] = se[v];
            s_red[wave][2][r] = sa[v];
        }
    __syncthreads();
    if (tid < 48) {
        int s = tid >> 4, r = tid & 15;
        float m = 0.f;
        for (int w = 0; w < 8; ++w) m += s_red[w][s][r];
        s_fin[1][s][r] = m;
    }
    __syncthreads();
    float ish[8], ise[8], isa[8];
    #pragma unroll
    for (int v = 0; v < 8; ++v) {           // sums >= 1, safe to invert
        int r = v + 8 * half;
        ish[v] = 1.f / s_fin[1][0][r];
        ise[v] = 1.f / s_fin[1][1][r];
        isa[v] = 1.f / s_fin[1][2][r];
    }

    // ================= pass C: fused A_dyn -> out =================
    for (int jt = wave; jt < NN / 16; jt += 8) {
        int j0 = jt * 16;
        v8f ch, ca; float ev[8], bv[8];
        tile_scores(j0, ch, ca, ev, bv);
        #pragma unroll
        for (int v = 0; v < 8; ++v) {
            float Ah = __expf(ch[v] - gmh[v]) * ish[v];
            float Ae = eflag * __expf(ev[v] - gme[v]) * ise[v];
            float Aa = __expf(ca[v] - gma[v]) * isa[v];
            float Ad = 0.5f * (gamma * Ah + (1.f - gamma) * Ae)
                     + 0.5f * (bv[v] + Aa);
            out[(size_t)(i0 + v + 8 * half) * NN + j0 + c] = Ad;
        }
    }
}

// ---------------- exact per-row top-20 + identity + degree -----------------
__global__ __launch_bounds__(256) void dg_kT(float* __restrict__ out,
                                             float* __restrict__ deg) {
    __shared__ float row[NN];
    __shared__ float rv[8];
    __shared__ int   ri[8];
    __shared__ float selv[TOPK];
    __shared__ int   seli[TOPK];
    __shared__ float bcastv;
    __shared__ int   bcasti;

    const int i   = blockIdx.x;
    const int tid = threadIdx.x;
    float* orow = out + (size_t)i * NN;

    for (int k = 0; k < 4; ++k)
        ((float4*)row)[k * 256 + tid] = ((const float4*)orow)[k * 256 + tid];
    __syncthreads();

    // local candidate over contiguous chunk row[tid*16 .. +15]
    float lv; int li;
    auto rescan = [&]() {
        lv = -1.f; li = tid * 16;               // A_dyn >= 0; removed = -2
        #pragma unroll
        for (int k = 0; k < 16; ++k) {
            int idx = tid * 16 + k;
            float x = row[idx];
            if (x > lv || (x == lv && idx < li)) { lv = x; li = idx; }
        }
    };
    rescan();

    for (int it = 0; it < TOPK; ++it) {
        float v = lv; int ix = li;
        for (int off = 16; off > 0; off >>= 1) {   // wave32 argmax
            float ov = __shfl_xor(v, off, 32);
            int   oi = __shfl_xor(ix, off, 32);
            if (ov > v || (ov == v && oi < ix)) { v = ov; ix = oi; }
        }
        if ((tid & 31) == 0) { rv[tid >> 5] = v; ri[tid >> 5] = ix; }
        __syncthreads();
        if (tid == 0) {
            float bv = rv[0]; int bi = ri[0];
            for (int w = 1; w < 8; ++w)
                if (rv[w] > bv || (rv[w] == bv && ri[w] < bi)) { bv = rv[w]; bi = ri[w]; }
            selv[it] = bv; seli[it] = bi; bcastv = bv; bcasti = bi;
        }
        __syncthreads();
        int widx = bcasti;
        if ((widx >> 4) == tid) { row[widx] = -2.f; rescan(); }
        __syncthreads();
    }

    // rebuild: zeros + selected values + identity
    for (int k = 0; k < 16; ++k) row[tid * 16 + k] = 0.f;
    __syncthreads();
    if (tid < TOPK) row[seli[tid]] = selv[tid];
    __syncthreads();
    if (tid == 0) {
        float d = 1.0f;
        for (int k = 0; k < TOPK; ++k) d += selv[k];
        deg[i] = d;
        row[i] += 1.0f;                           // A = A_dyn_masked + I
    }
    __syncthreads();
    for (int k = 0; k < 4; ++k)
        ((float4*)orow)[k * 256 + tid] = ((const float4*)row)[k * 256 + tid];
}

// ---------------- symmetric normalization in place -------------------------
__global__ __launch_bounds__(256) void dg_kS(float* __restrict__ out,
                                             const float* __restrict__ deg) {
    const int i = blockIdx.x;
    float di = rsqrtf(deg[i] + EPS_F);            // deg >= 1, never inf
    float* orow = out + (size_t)i * NN;
    for (int k = 0; k < 16; ++k) {
        int j = k * 256 + threadIdx.x;
        float dj = rsqrtf(deg[j] + EPS_F);
        orow[j] *= di * dj;
    }
}

extern "C" void kernel_launch(void* const* d_in, const int* in_sizes, int n_in,
                              void* d_out, int out_size, void* d_ws, size_t ws_size,
                              hipStream_t stream) {
    const float* x_hist = (const float*)d_in[0];   // [16,12,N,2]
    const float* event  = (const float*)d_in[1];   // [16,12,N,1]
    const float* base   = (const float*)d_in[2];   // [N,N]
    const float* nv1    = (const float*)d_in[3];   // [N,10]
    const float* nv2    = (const float*)d_in[4];   // [10,N]
    const float* alpha  = (const float*)d_in[5];   // scalar
    float* out = (float*)d_out;

    // ws layout (floats): [0,2N) feat | [2N,3N) sev | [3N] flag | [3N+256,+N) deg
    float* wsf  = (float*)d_ws;
    float* feat = wsf;
    float* sev  = wsf + 2 * NN;
    int*   flag = (int*)(wsf + 3 * NN);
    float* deg  = wsf + 3 * NN + 256;

    dg_kZ<<<(3 * NN + 256 + 255) / 256, 256, 0, stream>>>(wsf);
    dg_kP1<<<dim3(NN / 256, 8), 256, 0, stream>>>(x_hist, event, feat, sev);
    dg_kP2<<<NN / 256, 256, 0, stream>>>(feat, sev, flag);
    dg_kW<<<NN / 16, 256, 0, stream>>>(base, nv1, nv2, feat, sev, flag, alpha, out);
    dg_kT<<<NN, 256, 0, stream>>>(out, deg);
    dg_kS<<<NN, 256, 0, stream>>>(out, deg);
}